// CalibrationLoss_48258252538340
// MI455X (gfx1250) — compile-verified
//
#include <hip/hip_runtime.h>
#include <hip/hip_bf16.h>

#define N_BINS 15
#define SLOTS 30        // slots 0..14 = conf sums, 15..29 = acc sums
#define SLOTS_PAD 32    // padded so the WMMA epilogue reads zeros for 30,31
#define TPB 256
#define MAIN_BLOCKS 2048

typedef float f4  __attribute__((ext_vector_type(4)));
typedef float v2f __attribute__((ext_vector_type(2)));
typedef float v8f __attribute__((ext_vector_type(8)));

__global__ void calib_zero_ws_kernel(float* __restrict__ ws) {
    if (threadIdx.x < SLOTS_PAD) ws[threadIdx.x] = 0.0f;
}

__device__ __forceinline__ void calib_accum_one(float g, float a, float b, float t,
                                                float* hist, int tid) {
    const float d = a - 1.0f + 1e-8f;
    const float conf = d / (d + b);                    // == 1/(1 + b/d)
    const float err = fabsf(t - g);
    const float acc = 1.0f - fminf(err * (1.0f / 2.0f), 1.0f); // err >= 0
    int bin = (int)floorf(conf * (float)N_BINS);
    bin = bin < 0 ? 0 : (bin > (N_BINS - 1) ? (N_BINS - 1) : bin);
    const float w = (conf >= 0.0f && conf <= 1.0f) ? 1.0f : 0.0f;
    // Per-lane private slots; 256 % 64banks == 0 -> bank = (base+lane) % 64,
    // independent of the data-dependent bin: conflict-free ds_add_f32.
    atomicAdd(&hist[bin * TPB + tid], conf * w);
    atomicAdd(&hist[(bin + N_BINS) * TPB + tid], acc * w);
}

__global__ __launch_bounds__(TPB) void calib_hist_kernel(
    const float* __restrict__ gamma, const float* __restrict__ alpha,
    const float* __restrict__ beta, const float* __restrict__ targets,
    float* __restrict__ ws, int n) {
    __shared__ float hist[SLOTS_PAD * TPB];   // 32 KB
    const int tid = threadIdx.x;

#pragma unroll
    for (int s = 0; s < SLOTS_PAD; ++s) hist[s * TPB + tid] = 0.0f;
    __syncthreads();

    // ---- streaming main loop: 4x b128 NT loads per iteration ----
    const int n4 = n >> 2;
    const f4* __restrict__ gp = (const f4*)gamma;
    const f4* __restrict__ ap = (const f4*)alpha;
    const f4* __restrict__ bp = (const f4*)beta;
    const f4* __restrict__ tp = (const f4*)targets;

    for (int i = blockIdx.x * TPB + tid; i < n4; i += gridDim.x * TPB) {
        const f4 g = __builtin_nontemporal_load(&gp[i]);
        const f4 a = __builtin_nontemporal_load(&ap[i]);
        const f4 b = __builtin_nontemporal_load(&bp[i]);
        const f4 t = __builtin_nontemporal_load(&tp[i]);
#pragma unroll
        for (int c = 0; c < 4; ++c)
            calib_accum_one(g[c], a[c], b[c], t[c], hist, tid);
    }
    // scalar tail (N divisible by 4 for the reference, but stay general)
    for (int i = (n4 << 2) + blockIdx.x * TPB + tid; i < n; i += gridDim.x * TPB)
        calib_accum_one(gamma[i], alpha[i], beta[i], targets[i], hist, tid);
    __syncthreads();

    // ---- tree-reduce 256 copies -> 4 copies per slot ----
    for (int stride = TPB / 2; stride >= 4; stride >>= 1) {
        if (tid < stride) {
#pragma unroll
            for (int s = 0; s < SLOTS_PAD; ++s)
                hist[s * TPB + tid] += hist[s * TPB + tid + stride];
        }
        __syncthreads();
    }

    // ---- WMMA epilogue: row-sum the remaining 16x4 tiles with B = ones ----
    // A layout (f32 16x4): lanes 0-15 -> {K0,K1}, lanes 16-31 -> {K2,K3}.
    // All waves execute (EXEC all ones as WMMA requires); only wave0 writes.
    const int lane = tid & 31;
    const int m = lane & 15;
    const int kb = (lane < 16) ? 0 : 2;
    v2f a0, a1;
    a0.x = hist[m * TPB + kb];
    a0.y = hist[m * TPB + kb + 1];
    a1.x = hist[(16 + m) * TPB + kb];        // slots 16..31 (30,31 are zero pad)
    a1.y = hist[(16 + m) * TPB + kb + 1];
    const v2f bones = {1.0f, 1.0f};
    v8f c0 = {}, c1 = {};
    c0 = __builtin_amdgcn_wmma_f32_16x16x4_f32(false, a0, false, bones,
                                               (short)0, c0, false, false);
    c1 = __builtin_amdgcn_wmma_f32_16x16x4_f32(false, a1, false, bones,
                                               (short)0, c1, false, false);
    // D layout: lane 0 VGPR r = rowsum(r); lane 16 VGPR r = rowsum(8+r).
    if (tid == 0 || tid == 16) {
        const int rbase = (tid == 0) ? 0 : 8;
#pragma unroll
        for (int r = 0; r < 8; ++r) {
            unsafeAtomicAdd(&ws[rbase + r], c0[r]);          // slots 0..15
            const int s1 = 16 + rbase + r;                   // slots 16..31
            if (s1 < SLOTS) unsafeAtomicAdd(&ws[s1], c1[r]);
        }
    }
}

__global__ void calib_finalize_kernel(const float* __restrict__ ws,
                                      float* __restrict__ out, float inv_total) {
    if (threadIdx.x == 0) {
        float s = 0.0f;
#pragma unroll
        for (int b = 0; b < N_BINS; ++b)
            s += fabsf(ws[b] - ws[b + N_BINS]);
        out[0] = s * inv_total;
    }
}

extern "C" void kernel_launch(void* const* d_in, const int* in_sizes, int n_in,
                              void* d_out, int out_size, void* d_ws, size_t ws_size,
                              hipStream_t stream) {
    const float* gamma   = (const float*)d_in[0];
    const float* alpha   = (const float*)d_in[1];
    const float* beta    = (const float*)d_in[2];
    const float* targets = (const float*)d_in[3];
    float* ws  = (float*)d_ws;
    float* out = (float*)d_out;
    const int n = in_sizes[0];

    calib_zero_ws_kernel<<<1, 64, 0, stream>>>(ws);
    calib_hist_kernel<<<MAIN_BLOCKS, TPB, 0, stream>>>(gamma, alpha, beta,
                                                       targets, ws, n);
    calib_finalize_kernel<<<1, 32, 0, stream>>>(ws, out, 1.0f / (float)n);
}